// MSDSNv3torch_67422396612979
// MI455X (gfx1250) — compile-verified
//
#include <hip/hip_runtime.h>

typedef float v2f __attribute__((ext_vector_type(2)));
typedef float v8f __attribute__((ext_vector_type(8)));

#define TILE_W 32

__device__ inline v8f wmma_f32_16x16x4(v2f a, v2f b, v8f c) {
  // 8 args: (neg_a, A, neg_b, B, c_mod, C, reuse_a, reuse_b)
  return __builtin_amdgcn_wmma_f32_16x16x4_f32(false, a, false, b, (short)0, c,
                                               false, false);
}

// x:      (2, 64, 256, 256)
// y:      (4, 64, 256, 256)
// offpw:  (4, 16, 16, 3, 3)   off_w: (4, 18, 64)   off_b: (4, 18)
// maskpw: (16, 16, 3, 3)      mask_w: (9, 64)      mask_b: (9,)
// out:    (4, 64, 9, 256, 256)
__global__ __launch_bounds__(256) void msdsn_fused(
    const float* __restrict__ x, const float* __restrict__ y,
    const float* __restrict__ offpw, const float* __restrict__ offw,
    const float* __restrict__ offb, const float* __restrict__ maskpw,
    const float* __restrict__ maskw, const float* __restrict__ maskb,
    float* __restrict__ out) {
  __shared__ float s_xlo[16 * 3 * 34];       // x[0:16] halo: [ch][row 0..2][col 0..33]
  __shared__ float s_xhi[48 * TILE_W];       // x[16:64] at row h
  __shared__ float s_z1[5 * 16 * TILE_W];    // pconv outputs, 5 heads
  __shared__ float s_off[4 * 18 * TILE_W];   // offsets per group
  __shared__ float s_mask[9 * TILE_W];       // mask

  const int tid = threadIdx.x;
  const int wt = blockIdx.x & 7;             // strip index (W/32 = 8)
  const int h = (blockIdx.x >> 3) & 255;     // row
  const int bb = blockIdx.x >> 11;           // batch of x (0..1)
  const int w0 = wt * TILE_W;

  // ---------------- Phase A0: stage x into LDS (zero-padded halo) -----------
  for (int idx = tid; idx < 16 * 3 * 34; idx += 256) {
    int ch = idx / 102;
    int rem = idx % 102;
    int r = rem / 34;
    int c = rem % 34;
    int gr = h - 1 + r;
    int gc = w0 - 1 + c;
    float v = 0.f;
    if (gr >= 0 && gr < 256 && gc >= 0 && gc < 256)
      v = x[((bb * 64 + ch) * 256 + gr) * 256 + gc];
    s_xlo[idx] = v;
  }
  for (int idx = tid; idx < 48 * TILE_W; idx += 256) {
    int ch = idx / TILE_W + 16;
    int c = idx % TILE_W;
    s_xhi[idx] = x[((bb * 64 + ch) * 256 + h) * 256 + w0 + c];
  }
  __syncthreads();

  const int wv = tid >> 5;     // wave id (8 waves)
  const int lane = tid & 31;   // lane in wave32
  const int lm = lane & 15;
  const int hi = lane >> 4;

  // ------------- Phase A1: 3x3 pconv (16->16) via WMMA f32 16x16x4 ----------
  // Tasks: head (4 offset heads + 1 mask head) x 2 pixel-subtiles = 10 tasks.
  for (int t = wv; t < 10; t += 8) {
    int hd = t >> 1;           // uniform per wave
    int st = t & 1;
    const float* wp = (hd < 4) ? (offpw + hd * 16 * 16 * 9) : maskpw;
    v8f acc = {0.f, 0.f, 0.f, 0.f, 0.f, 0.f, 0.f, 0.f};
    for (int kh = 0; kh < 3; ++kh)
      for (int kw = 0; kw < 3; ++kw)
#pragma unroll
        for (int kc = 0; kc < 4; ++kc) {
          v2f a, b;
#pragma unroll
          for (int j = 0; j < 2; ++j) {
            int ch = 4 * kc + j + 2 * hi;                       // A: K index
            a[j] = wp[(lm * 16 + ch) * 9 + kh * 3 + kw];        // W[oc=lm][ic=ch]
            b[j] = s_xlo[ch * 102 + kh * 34 + st * 16 + lm + kw];
          }
          acc = wmma_f32_16x16x4(a, b, acc);
        }
#pragma unroll
    for (int r = 0; r < 8; ++r)
      s_z1[hd * 16 * TILE_W + (r + 8 * hi) * TILE_W + st * 16 + lm] = acc[r];
  }
  __syncthreads();

  // ------- Phase A2: 1x1 convs (64->18 offsets, 64->9 mask) via WMMA --------
  // Tasks: 4 groups x 2 out-row-tiles x 2 subtiles = 16, + mask x 2 = 18.
  for (int t = wv; t < 18; t += 8) {
    const bool ismask = (t >= 16);
    int hd = ismask ? 4 : (t >> 2);
    int ctile = ismask ? 0 : ((t >> 1) & 1);
    int st = t & 1;
    int px = st * 16 + lm;
    int oc_a = ctile * 16 + lm;
    v8f acc = {0.f, 0.f, 0.f, 0.f, 0.f, 0.f, 0.f, 0.f};
#pragma unroll 4
    for (int kc = 0; kc < 16; ++kc) {
      v2f a, b;
#pragma unroll
      for (int j = 0; j < 2; ++j) {
        int ch = 4 * kc + j + 2 * hi;  // 0..63
        float av = 0.f;
        if (ismask) {
          if (oc_a < 9) av = maskw[oc_a * 64 + ch];
        } else {
          if (oc_a < 18) av = offw[(hd * 18 + oc_a) * 64 + ch];
        }
        a[j] = av;
        b[j] = (ch < 16) ? s_z1[hd * 16 * TILE_W + ch * TILE_W + px]
                         : s_xhi[(ch - 16) * TILE_W + px];
      }
      acc = wmma_f32_16x16x4(a, b, acc);
    }
#pragma unroll
    for (int r = 0; r < 8; ++r) {
      int oc = ctile * 16 + r + 8 * hi;
      if (ismask) {
        if (oc < 9) s_mask[oc * TILE_W + px] = acc[r] + maskb[oc];
      } else if (oc < 18) {
        int k = oc >> 1;
        float pd = (oc & 1) ? (float)(k % 3 - 1) : (float)(k / 3 - 1);
        s_off[(hd * 18 + oc) * TILE_W + px] =
            acc[r] + offb[hd * 18 + oc] + pd * (float)(2 * hd + 1);
      }
    }
  }
  __syncthreads();

  // --------- Phase B: bilinear warp + mask, fully-coalesced NT stores -------
  const int pxl = tid & 31;       // pixel in strip
  const int sub = tid >> 5;       // 0..7
  const int s = sub >> 2;         // y-half (y1 / y2)
  const int cq = sub & 3;         // channel quartet within group
  const int wg = w0 + pxl;        // global column
  for (int i = 0; i < 4; ++i)
    for (int k = 0; k < 9; ++k) {
      float ox = s_off[(i * 18 + 2 * k) * TILE_W + pxl];
      float oy = s_off[(i * 18 + 2 * k + 1) * TILE_W + pxl];
      float pxf = fminf(fmaxf((float)wg + ox, 0.f), 255.f);
      float pyf = fminf(fmaxf((float)h + oy, 0.f), 255.f);
      float x0f = floorf(pxf), y0f = floorf(pyf);
      float wx = pxf - x0f, wy = pyf - y0f;
      int x0 = (int)x0f, y0 = (int)y0f;
      int x1 = min(x0 + 1, 255), y1 = min(y0 + 1, 255);
      float mk = s_mask[k * TILE_W + pxl];
      float w00 = (1.f - wx) * (1.f - wy) * mk;
      float w01 = wx * (1.f - wy) * mk;
      float w10 = (1.f - wx) * wy * mk;
      float w11 = wx * wy * mk;
      int i00 = y0 * 256 + x0, i01 = y0 * 256 + x1;
      int i10 = y1 * 256 + x0, i11 = y1 * 256 + x1;
#pragma unroll
      for (int cg = 0; cg < 4; ++cg) {
        int ch = i * 16 + cq * 4 + cg;
        const float* yp = y + (size_t)((s * 2 + bb) * 64 + ch) * 65536u;
        float v = yp[i00] * w00 + yp[i01] * w01 + yp[i10] * w10 + yp[i11] * w11;
        // Output is write-once streamed (604 MB): non-temporal store so the
        // store stream does not evict the L2-resident gather source y (67 MB).
        __builtin_nontemporal_store(
            v, out + (size_t)(((s * 2 + bb) * 64 + ch) * 9 + k) * 65536u +
                   h * 256 + wg);
      }
    }
}

extern "C" void kernel_launch(void* const* d_in, const int* in_sizes, int n_in,
                              void* d_out, int out_size, void* d_ws,
                              size_t ws_size, hipStream_t stream) {
  (void)in_sizes; (void)n_in; (void)d_ws; (void)ws_size; (void)out_size;
  const float* x = (const float*)d_in[0];
  const float* y = (const float*)d_in[1];
  const float* offpw = (const float*)d_in[2];
  const float* offw = (const float*)d_in[3];
  const float* offb = (const float*)d_in[4];
  const float* maskpw = (const float*)d_in[5];
  const float* maskw = (const float*)d_in[6];
  const float* maskb = (const float*)d_in[7];
  // grid = b(2) * H(256) * (W/TILE_W = 8) = 4096 workgroups of 256 threads
  msdsn_fused<<<dim3(4096), dim3(256), 0, stream>>>(
      x, y, offpw, offw, offb, maskpw, maskw, maskb, (float*)d_out);
}